// NoiseBandSynth_61744449847516
// MI455X (gfx1250) — compile-verified
//
#include <hip/hip_runtime.h>

typedef float v2f __attribute__((ext_vector_type(2)));
typedef float v8f __attribute__((ext_vector_type(8)));

// Problem constants (from the reference)
constexpr int T_OUT   = 8192;   // output samples per batch
constexpr int BAND    = 4096;   // noiseband length (t mod BAND)
constexpr int NBANDS  = 2048;   // reduction dim (K)
constexpr int NFRAMES = 256;    // control-rate frames

// Window blocks: block v covers one frame period of output time so that up to
// TWO 16-wide t-tiles share a single 16-column frame set (4 batches x
// {fA0,fA1,fB0,fB1}), amortizing the scattered amplitude (B) gather across
// 2 WMMA tiles while nb (A) columns stream coalesced and are read once.
//   v = 0        : 1 tile,  t0 = 0      (clamped region, i0 = 0, w = 0)
//   v = 1..127   : 2 tiles, t0 = 32v-16 (i0 = v-1 for the whole window)
//   v = 128      : 1 tile,  t0 = 4080   (i0 = 127; h=1 side clamps at 255)
// NT is compile-time so the inner loop is branch-free and pipelines.
// 512 threads = 16 waves split K = 2048 into 128-deep slices.
template <int NT>
__global__ __launch_bounds__(512) void nbsynth_wmma_kernel(
    const float* __restrict__ amp,   // [4, 2048, 256]
    const float* __restrict__ nb,    // [2048, 4096]
    float* __restrict__ out)         // [4, 8192]
{
    __shared__ float lds[16 * 512];  // 16 waves x (2 tiles x 16 x 16) partials

    const int v = (NT == 2) ? ((int)blockIdx.x + 1)
                            : ((blockIdx.x == 0) ? 0 : 128);
    const int tid   = threadIdx.x;
    const int wave  = tid >> 5;
    const int lane  = tid & 31;
    const int lhalf = lane >> 4;     // K sub-pair select per 16x4 f32 layout
    const int lmod  = lane & 15;

    const int t0 = (v == 0) ? 0 : (32 * v - 16);   // first output t (first half)

    // Shared frame set for the whole window (both tiles, both halves).
    const int fA0 = (v == 0) ? 0 : (v - 1);
    const int fA1 = min(fA0 + 1, NFRAMES - 1);
    const int fB0 = min(v + 127, NFRAMES - 1);
    const int fB1 = min(fB0 + 1, NFRAMES - 1);
    int fr[4] = { fA0, fA1, fB0, fB1 };

    // This wave's K slice: rows (k0, k0+1) in v0/v1, lanes 16-31 get (+2, +3).
    const int k0 = wave * 128 + lhalf * 2;

    // A fragments: A_j[m][k] = nb[k][t0 + 16j + m]   (coalesced over t)
    const float* aptr0 = nb + k0 * BAND + t0 + lmod;
    const float* aptr1 = aptr0 + 16;
    // B fragment (shared by both tiles): B[k][c] = amp[b(c)][k][frame(c)]
    const int bidx  = lmod >> 2;
    const int frame = fr[lmod & 3];
    const float* bptr = amp + (bidx * NBANDS + k0) * NFRAMES + frame;

    v8f c0 = {};
    v8f c1 = {};
    #pragma unroll 4
    for (int it = 0; it < 32; ++it) {
        v2f b;
        b.x = bptr[0];
        b.y = bptr[NFRAMES];
        v2f a0;
        a0.x = aptr0[0];
        a0.y = aptr0[BAND];
        __builtin_prefetch(aptr0 + 16 * BAND, 0, 1);   // stream hint for nb
        c0 = __builtin_amdgcn_wmma_f32_16x16x4_f32(
                 false, a0, false, b, (short)0, c0, false, false);
        if (NT == 2) {
            v2f a1;
            a1.x = aptr1[0];
            a1.y = aptr1[BAND];
            c1 = __builtin_amdgcn_wmma_f32_16x16x4_f32(
                     false, a1, false, b, (short)0, c1, false, false);
        }
        aptr0 += 4 * BAND;
        aptr1 += 4 * BAND;
        bptr  += 4 * NFRAMES;
    }

    // Spill partial tiles: lane holds rows m = vv + 8*lhalf, col = lmod.
    #pragma unroll
    for (int vv = 0; vv < 8; ++vv) {
        const int m = vv + 8 * lhalf;
        lds[wave * 512 + m * 16 + lmod] = c0[vv];
        if (NT == 2) lds[wave * 512 + 256 + m * 16 + lmod] = c1[vv];
    }
    __syncthreads();

    // Reduce the 16 per-wave partials. Slot j is only touched by thread
    // (j & 511), so the in-place write needs no extra barrier.
    if (tid < NT * 256) {
        float s = 0.0f;
        #pragma unroll
        for (int w = 0; w < 16; ++w) s += lds[w * 512 + tid];
        lds[tid] = s;
    }
    __syncthreads();

    // Epilogue: NT*128 outputs = NT tiles x 16 t x 4 batches x 2 halves.
    if (tid < NT * 128) {
        const int j = tid >> 7;
        const int r = tid & 127;
        const int m = r & 15;
        const int b = (r >> 4) & 3;
        const int h = r >> 6;
        const int t = t0 + 16 * j + m + h * BAND;

        float x = (float)t * 0.03125f + 0.015625f - 0.5f;
        x = fminf(fmaxf(x, 0.0f), 255.0f);
        const float w = x - floorf(x);

        const int cbase = b * 4 + 2 * h;
        const float p0 = lds[j * 256 + m * 16 + cbase];
        const float p1 = lds[j * 256 + m * 16 + cbase + 1];
        out[b * T_OUT + t] = p0 * (1.0f - w) + p1 * w;
    }
}

extern "C" void kernel_launch(void* const* d_in, const int* in_sizes, int n_in,
                              void* d_out, int out_size, void* d_ws, size_t ws_size,
                              hipStream_t stream) {
    const float* amp = (const float*)d_in[0];   // [4, 2048, 256] f32
    const float* nb  = (const float*)d_in[1];   // [2048, 4096] f32
    float* out = (float*)d_out;                 // [4, 1, 8192] f32

    // 127 interior two-tile windows (v = 1..127)
    hipLaunchKernelGGL((nbsynth_wmma_kernel<2>), dim3(127), dim3(512), 0, stream,
                       amp, nb, out);
    // 2 one-tile edge windows (v = 0 and v = 128); disjoint outputs
    hipLaunchKernelGGL((nbsynth_wmma_kernel<1>), dim3(2), dim3(512), 0, stream,
                       amp, nb, out);
}